// IE_23991687315466
// MI455X (gfx1250) — compile-verified
//
#include <hip/hip_runtime.h>
#include <math.h>

typedef __attribute__((ext_vector_type(2))) float v2f;
typedef __attribute__((ext_vector_type(8))) float v8f;

#define NROWS 4095          // feature_matrix rows (= masks 1..4095)
#define NCOLS 8190          // feature_matrix cols (= 2*n)
#define NSLICES 16
#define ITERS_TOTAL 2048    // ceil(8190/4)
#define ITERS_PER_SLICE (ITERS_TOTAL / NSLICES)   // 128

// ---------------------------------------------------------------------------
// Kernel 1: lattice DP for FM (one workgroup, g[] in LDS, barrier per level).
// g indexed by mask m (1..4095). Level k reads only level k-1 values.
// FMbuf[m-1] = min(g[m],1), g[4095] forced to 1.0; FMbuf[4095] = 0 (pad so
// the matvec k-tail multiplies out-of-range columns by exactly zero).
// ---------------------------------------------------------------------------
__global__ void ie_build_fm(const float* __restrict__ ie, float* __restrict__ FMbuf) {
  __shared__ float g[4096];
  const int tid = threadIdx.x;                       // blockDim = 1024
  for (int k = 1; k <= 11; ++k) {
    for (int m = 1 + tid; m < 4096; m += 1024) {
      if (__popc(m) == k) {
        float val = fabsf(ie[m - 1]);                // ie has 4094 entries; m<=4094 here
        if (k == 1) {
          g[m] = val;
        } else {
          float best = -INFINITY;
          int t = m;
          while (t) {
            int bset = t & (-t);
            best = fmaxf(best, g[m ^ bset]);         // child: drop one set bit
            t ^= bset;
          }
          g[m] = best + val;
        }
      }
    }
    __syncthreads();
  }
  for (int m = 1 + tid; m < 4096; m += 1024) {
    float gg = (m == 4095) ? 1.0f : g[m];
    FMbuf[m - 1] = fminf(gg, 1.0f);
  }
  if (tid == 0) FMbuf[4095] = 0.0f;
}

// ---------------------------------------------------------------------------
// One WMMA accumulation step: A tile element pair from *ap, B column 0 = ±fm.
// ---------------------------------------------------------------------------
__device__ __forceinline__ v8f ie_wmma_step(v8f c, const float* __restrict__ ap,
                                            float fm, bool col0) {
  v2f a = *(const v2f*)ap;
  v2f b;
  b.x = col0 ?  fm : 0.0f;
  b.y = col0 ? -fm : 0.0f;
  return __builtin_amdgcn_wmma_f32_16x16x4_f32(false, a, false, b,
                                               (short)0, c, false, false);
}

// ---------------------------------------------------------------------------
// Kernel 2: w_part[slice][r] = sum over slice's k-range of A[r,j]*s(j),
// s(2A)=FM[A], s(2A+1)=-FM[A].  WMMA f32 16x16x4, B nonzero only in column 0:
//   lane L (L&15==0): b = (s[k0+koff], s[k0+koff+1]),  koff = (L>>4)*2
// A tile per ISA layout: lane L loads A[rg*16 + (L&15), k0+koff .. +1].
// Column 0 of C accumulates the 16 row-dots (lane 0: M=0..7, lane 16: M=8..15).
// Only the final iteration of the final slice touches the k-tail; it is
// peeled out so the hot loop uses pure base+immediate-offset addressing.
// ---------------------------------------------------------------------------
__global__ void ie_matvec_wmma(const float* __restrict__ A,
                               const float* __restrict__ FMbuf,
                               float* __restrict__ w_part) {
  const int lane  = threadIdx.x;                     // blockDim = 32 (one wave)
  const int rg    = blockIdx.x >> 4;                 // 0..255 row groups
  const int slice = blockIdx.x & (NSLICES - 1);      // 0..15 k slices
  int row = rg * 16 + (lane & 15);
  if (row > NROWS - 1) row = NROWS - 1;              // row 4095 doesn't exist; clamp load
  const int  grp  = lane >> 4;                       // 0 or 1
  const int  koff = grp << 1;                        // 0 or 2
  const bool col0 = (lane & 15) == 0;
  const float* Arow = A + (size_t)row * NCOLS;
  const float* ap = Arow + slice * ITERS_PER_SLICE * 4 + koff;
  const float* fp = FMbuf + slice * ITERS_PER_SLICE * 2 + grp;

  v8f c = {0.f, 0.f, 0.f, 0.f, 0.f, 0.f, 0.f, 0.f};

  if (slice != NSLICES - 1) {                        // wave-uniform branch
#pragma unroll 8
    for (int it = 0; it < ITERS_PER_SLICE; ++it)
      c = ie_wmma_step(c, ap + it * 4, fp[it * 2], col0);
  } else {
#pragma unroll 4
    for (int it = 0; it < ITERS_PER_SLICE - 1; ++it)
      c = ie_wmma_step(c, ap + it * 4, fp[it * 2], col0);
    // tail: k0 = 8188. grp0 reads cols 8188/8189; grp1 cols would be
    // 8190/8191 (OOB) -> read a safe dummy address, FMbuf[4095]==0 zeroes it.
    float fm = FMbuf[4094 + grp];
    c = ie_wmma_step(c, grp ? Arow : (Arow + 8188), fm, col0);
  }

  if (col0) {
    // w_part rows are padded to 4096; slot r==4095 may hold garbage but the
    // reducer never reads it, so store unconditionally as two b128s.
    float* dst = w_part + slice * 4096 + rg * 16 + (grp << 3);
    *(float4*)(dst)     = make_float4(c[0], c[1], c[2], c[3]);
    *(float4*)(dst + 4) = make_float4(c[4], c[5], c[6], c[7]);
  }
}

// ---------------------------------------------------------------------------
// Kernel 3: fixed-order slice reduction -> w[r]   (deterministic, no atomics)
// ---------------------------------------------------------------------------
__global__ void ie_reduce_w(const float* __restrict__ w_part, float* __restrict__ w) {
  int r = blockIdx.x * blockDim.x + threadIdx.x;
  if (r < NROWS) {
    float s = 0.0f;
#pragma unroll
    for (int sl = 0; sl < NSLICES; ++sl) s += w_part[sl * 4096 + r];
    w[r] = s;
  }
}

// ---------------------------------------------------------------------------
// Kernel 4: per-batch-row subset-product dots. One wave per row.
// mask m = h*32 + lane; P(m) = Qhi(h over x[5..11]) * Plo(lane over x[0..4]).
// accl = Pl.w, accu = Pu.w; left = min(accl,accu), right = accu.
// ---------------------------------------------------------------------------
__global__ void ie_batch_dot(const float* __restrict__ x,
                             const float* __restrict__ w,
                             float* __restrict__ out) {
  const int wid  = threadIdx.x >> 5;                 // blockDim = 256 -> 8 waves
  const int lane = threadIdx.x & 31;
  const int b    = blockIdx.x * 8 + wid;             // grid 512 -> 4096 rows
  const float* xr = x + b * 24;
  float lo[12], hi[12];
#pragma unroll
  for (int i = 0; i < 12; ++i) { lo[i] = xr[i]; hi[i] = xr[12 + i]; }

  float pl = 1.0f, pu = 1.0f;                        // low-5-bit products per lane
#pragma unroll
  for (int i = 0; i < 5; ++i)
    if ((lane >> i) & 1) { pl *= lo[i]; pu *= hi[i]; }

  float accl = 0.0f, accu = 0.0f;
  for (int h = 0; h < 128; ++h) {                    // h uniform across the wave
    float ql = 1.0f, qu = 1.0f;
#pragma unroll
    for (int i = 0; i < 7; ++i)
      if ((h >> i) & 1) { ql *= lo[5 + i]; qu *= hi[5 + i]; }
    int m = h * 32 + lane;
    float wv = (m > 0) ? w[m - 1] : 0.0f;            // skip empty mask
    accl += wv * (ql * pl);
    accu += wv * (qu * pu);
  }
#pragma unroll
  for (int off = 16; off > 0; off >>= 1) {           // wave32 reduction
    accl += __shfl_xor(accl, off, 32);
    accu += __shfl_xor(accu, off, 32);
  }
  if (lane == 0) {
    out[b]        = fminf(accl, accu);               // left
    out[4096 + b] = accu;                            // right
  }
}

// ---------------------------------------------------------------------------
extern "C" void kernel_launch(void* const* d_in, const int* in_sizes, int n_in,
                              void* d_out, int out_size, void* d_ws, size_t ws_size,
                              hipStream_t stream) {
  const float* x   = (const float*)d_in[0];   // (4096, 24)
  const float* ie  = (const float*)d_in[1];   // (4094, 1)
  const float* fmx = (const float*)d_in[2];   // (4095, 8190) row-major
  float* out = (float*)d_out;                 // 8192 floats: left | right
  float* ws  = (float*)d_ws;

  float* FMbuf  = ws;                         // 4096 floats (padded)
  float* w_part = ws + 4096;                  // 16 * 4096 floats
  float* w      = ws + 4096 + NSLICES * 4096; // 4096 floats

  ie_build_fm   <<<1, 1024, 0, stream>>>(ie, FMbuf);
  ie_matvec_wmma<<<256 * NSLICES, 32, 0, stream>>>(fmx, FMbuf, w_part);
  ie_reduce_w   <<<16, 256, 0, stream>>>(w_part, w);
  ie_batch_dot  <<<512, 256, 0, stream>>>(x, w, out);
}